// PureCartesianTensorProductO3Sparse_37967510897142
// MI455X (gfx1250) — compile-verified
//
#include <hip/hip_runtime.h>

// CDNA5 / gfx1250 wave32 WMMA tensor-product kernel.
// D[o,n] = sum_k W[o,k] * P[k,n],  k = a*16+b,  P[k,n] = A[n,a]*B[n,b]
// W  -> WMMA A-operand (16x4 per step) read from padded LDS rows
// P  -> WMMA B-operand (4x16 per step) built with 2 v_mul_f32 per step

typedef float v2f __attribute__((ext_vector_type(2)));
typedef float v8f __attribute__((ext_vector_type(8)));

#define NROW 258              // padded LDS row stride (dwords): even -> 8B-aligned b64, banks spread
#define PATH_LDS (16 * NROW)  // 4128 dwords per path

// One full K=256 contraction: 64 x V_WMMA_F32_16X16X4_F32.
// Wlane = per-lane LDS base: &smem[slot][o=lane&15][2*hi]
// a[16]: this lane's sample A-channel vector (same in both half-waves)
// b[16]: this lane's sample B-channel vector, rotated by 2*hi at load time
__device__ __forceinline__ void mm_path(v8f& acc, const float* Wlane,
                                        const float (&a)[16], const float (&b)[16]) {
#pragma unroll
  for (int ai = 0; ai < 16; ++ai) {
#pragma unroll
    for (int b0 = 0; b0 < 16; b0 += 4) {
      v2f wv = *(const v2f*)(Wlane + ai * 16 + b0);  // W[o][ai*16+b0 + 2*hi .. +1]
      v2f p;
      p[0] = a[ai] * b[b0];      // K = k0   (+2 in high half via rotated b[])
      p[1] = a[ai] * b[b0 + 1];  // K = k0+1 (+2 in high half)
      acc = __builtin_amdgcn_wmma_f32_16x16x4_f32(false, wv, false, p,
                                                  (short)0, acc, false, false);
    }
  }
}

// Cooperative copy of one 16x256 path weight into a padded LDS slot.
__device__ __forceinline__ void load_w_lds(float* dst, const float* __restrict__ src, int tid) {
  for (int idx = tid; idx < 4096; idx += 256) {
    int o = idx >> 8, k = idx & 255;
    dst[o * NROW + k] = src[idx];
  }
}

__global__ __launch_bounds__(256) void tp_o3_wmma_kernel(
    const float* __restrict__ x1, const float* __restrict__ x2,
    const float* __restrict__ wgt, float* __restrict__ out,
    int N, int ngroups) {
  __shared__ float smem[3 * PATH_LDS];  // 49,536 B

  const int tid   = threadIdx.x;
  const int lane  = tid & 31;
  const int wv    = tid >> 5;
  const int group = blockIdx.x * 8 + wv;
  const bool valid = group < ngroups;   // wave-uniform: EXEC stays all-1s inside
  const int hi  = lane >> 4;            // half-wave: supplies K+2 slices
  const int nl  = lane & 15;            // local sample / o-row
  const int rot = hi << 1;              // B-channel rotation for high half

  // ---- phase 1 weights: paths (0,0)->slot0, (0,1)->slot1, (1,0)->slot2
  load_w_lds(smem + 0 * PATH_LDS, wgt + 0 * 4096, tid);
  load_w_lds(smem + 1 * PATH_LDS, wgt + 1 * 4096, tid);
  load_w_lds(smem + 2 * PATH_LDS, wgt + 3 * 4096, tid);
  __syncthreads();

  int n = group * 16 + nl;
  if (n > N - 1) n = N - 1;  // tail clamp (N%16==0 for this workload; duplicates are benign)
  const float* x1n = x1 + (size_t)n * 416;
  const float* x2n = x2 + (size_t)n * 416;
  float* outn      = out + (size_t)n * 416;

  const float* Wl0 = smem + 0 * PATH_LDS + nl * NROW + rot;
  const float* Wl1 = smem + 1 * PATH_LDS + nl * NROW + rot;
  const float* Wl2 = smem + 2 * PATH_LDS + nl * NROW + rot;

  float a0[16], b0r[16];
  if (valid) {
#pragma unroll
    for (int j = 0; j < 16; ++j) a0[j] = x1n[j];                   // A L=0
#pragma unroll
    for (int j = 0; j < 16; ++j) b0r[j] = x2n[(j + rot) & 15];     // B L=0 (rotated)
  }

  if (valid) {
    // path (0,0): out0[o]
    v8f acc00 = {};
    mm_path(acc00, Wl0, a0, b0r);
#pragma unroll
    for (int r = 0; r < 8; ++r) outn[r + 8 * hi] = acc00[r];

    // out1[o,i] = path (0,1) + path (1,0)
    v8f acc1[3];
#pragma unroll
    for (int i = 0; i < 3; ++i) {
      acc1[i] = (v8f){};
      float br[16];
#pragma unroll
      for (int j = 0; j < 16; ++j) br[j] = x2n[16 + ((j + rot) & 15) * 3 + i];
      mm_path(acc1[i], Wl1, a0, br);
    }
#pragma unroll
    for (int i = 0; i < 3; ++i) {
      float as[16];
#pragma unroll
      for (int j = 0; j < 16; ++j) as[j] = x1n[16 + j * 3 + i];
      mm_path(acc1[i], Wl2, as, b0r);
    }
#pragma unroll
    for (int i = 0; i < 3; ++i)
#pragma unroll
      for (int r = 0; r < 8; ++r) outn[16 + (r + 8 * hi) * 3 + i] = acc1[i][r];

    // zero-fill columns 208..415 (harness poisons d_out); 16B-aligned float4 stores
    const float4 z = make_float4(0.f, 0.f, 0.f, 0.f);
    float4* zp = (float4*)(outn + 208 + hi * 104);
#pragma unroll
    for (int q = 0; q < 26; ++q) zp[q] = z;
  }

  __syncthreads();
  // ---- phase 2 weights: paths (0,2)->slot0, (1,1)->slot1, (2,0)->slot2
  load_w_lds(smem + 0 * PATH_LDS, wgt + 2 * 4096, tid);
  load_w_lds(smem + 1 * PATH_LDS, wgt + 4 * 4096, tid);
  load_w_lds(smem + 2 * PATH_LDS, wgt + 5 * 4096, tid);
  __syncthreads();

  if (valid) {
    v8f acc2[9];
#pragma unroll
    for (int t = 0; t < 9; ++t) acc2[t] = (v8f){};

    // path (0,2): out2[o,ij] += A0 x B2[:,ij]
#pragma unroll
    for (int ij = 0; ij < 9; ++ij) {
      float br[16];
#pragma unroll
      for (int j = 0; j < 16; ++j) br[j] = x2n[64 + ((j + rot) & 15) * 9 + ij];
      mm_path(acc2[ij], Wl0, a0, br);
    }
    // path (2,0): out2[o,ij] += A2[:,ij] x B0
#pragma unroll
    for (int ij = 0; ij < 9; ++ij) {
      float as[16];
#pragma unroll
      for (int j = 0; j < 16; ++j) as[j] = x1n[64 + j * 9 + ij];
      mm_path(acc2[ij], Wl2, as, b0r);
    }
    // path (1,1): out2[o,i*3+j2] += A1[:,i] x B1[:,j2]
#pragma unroll
    for (int i = 0; i < 3; ++i) {
      float as[16];
#pragma unroll
      for (int j = 0; j < 16; ++j) as[j] = x1n[16 + j * 3 + i];
#pragma unroll
      for (int j2 = 0; j2 < 3; ++j2) {
        float br[16];
#pragma unroll
        for (int j = 0; j < 16; ++j) br[j] = x2n[16 + ((j + rot) & 15) * 3 + j2];
        mm_path(acc2[i * 3 + j2], Wl1, as, br);
      }
    }
#pragma unroll
    for (int ij = 0; ij < 9; ++ij)
#pragma unroll
      for (int r = 0; r < 8; ++r) outn[64 + (r + 8 * hi) * 9 + ij] = acc2[ij][r];
  }
}

extern "C" void kernel_launch(void* const* d_in, const int* in_sizes, int n_in,
                              void* d_out, int out_size, void* d_ws, size_t ws_size,
                              hipStream_t stream) {
  const float* x1  = (const float*)d_in[0];
  const float* x2  = (const float*)d_in[1];
  const float* wgt = (const float*)d_in[2];
  float* out = (float*)d_out;

  const int N = in_sizes[0] / 416;          // 200000
  const int ngroups = (N + 15) / 16;        // 12500 (16 samples per wave)
  const int nblocks = (ngroups + 7) / 8;    // 8 waves per 256-thread block

  tp_o3_wmma_kernel<<<nblocks, 256, 0, stream>>>(x1, x2, wgt, out, N, ngroups);
}